// LinearMimo_9096740733411
// MI455X (gfx1250) — compile-verified
//
#include <hip/hip_runtime.h>
#include <stdint.h>

#define Bsz 32
#define Tsz 16384
#define Isz 8
#define Osz 8
#define NBt 3
#define NAt 2
#define TILE 16
#define NT (Tsz / TILE)   /* 1024 tiles */
#define BG 16             /* batches per group (WMMA N dim) */
#define NGRP (Bsz / BG)   /* 2 */

typedef float v2f __attribute__((ext_vector_type(2)));
typedef float v8f __attribute__((ext_vector_type(8)));
typedef unsigned int u32x4 __attribute__((ext_vector_type(4)));
typedef int i32x4 __attribute__((ext_vector_type(4)));
typedef int i32x8 __attribute__((ext_vector_type(8)));

__global__ __launch_bounds__(256) void mimo_iir_kernel(
    const float* __restrict__ bco, const float* __restrict__ aco,
    const float* __restrict__ u_in, const float* __restrict__ y0,
    const float* __restrict__ u0, float* __restrict__ yout)
{
  // u staging: [2 buffers][16 batches][18 times][8 inputs]  (TDM tile: 144 f32 x 16 rows)
  __shared__ float ubuf[2][BG * 18 * Isz];
  // per-wave 16x16 output tiles for the i-reduction
  __shared__ float ystage[8 * TILE * BG];
  // impulse-response coefficients h[0..16] per wave(channel)
  __shared__ float hsh[8][24];

  const int tid  = threadIdx.x;
  const int lane = tid & 31;
  const int w    = tid >> 5;            // wave id == input channel i
  const int i    = w;
  const int o    = blockIdx.x & 7;
  const int g    = blockIdx.x >> 3;     // batch group

  const int m     = lane & 15;               // WMMA M (rows of A) / N (cols of B)
  const int khalf = (lane < 16) ? 0 : 2;     // K offset of lane half (A/B layout)
  const int roff  = (lane < 16) ? 0 : 8;     // row offset of lane half (C/D layout)

  const int ch = o * Isz + i;
  const float b0 = bco[ch * NBt + 0], b1 = bco[ch * NBt + 1], b2 = bco[ch * NBt + 2];
  const float a0 = aco[ch * NAt + 0], a1 = aco[ch * NAt + 1];

  // ---- h sequence (per channel): h[0]=1, h[n]=-a0 h[n-1] - a1 h[n-2] ----
  if (lane == 0) {
    float hm1 = 1.f, hm2 = 0.f;
    hsh[w][0] = 1.f;
    for (int k = 1; k <= 16; ++k) {
      float hk = -a0 * hm1 - a1 * hm2;
      hsh[w][k] = hk;
      hm2 = hm1; hm1 = hk;
    }
  }

  // ---- preload tile 0 (needs u[-2],u[-1] from u0) with plain loads ----
  const int bbase = g * BG;
  for (int q = tid; q < BG * 18 * Isz; q += 256) {
    int b16 = q / (18 * Isz);
    int rem = q - b16 * (18 * Isz);
    int tl  = rem >> 3;          // 0..17  (time t = tl - 2)
    int ii  = rem & 7;
    int t   = tl - 2;
    float v;
    if (t >= 0) v = u_in[((size_t)(bbase + b16) * Tsz + t) * Isz + ii];
    else        v = u0[((bbase + b16) * Isz + ii) * NBt + (-1 - t)];
    ubuf[0][q] = v;
  }
  __syncthreads();

  // ---- A operand: lower-triangular Toeplitz H[m,K] = h[m-K], 4 K-slices of 16x4 ----
  v2f A[4];
#pragma unroll
  for (int s = 0; s < 4; ++s) {
#pragma unroll
    for (int j = 0; j < 2; ++j) {
      int K = 4 * s + j + khalf;
      int d = m - K;
      float hv = hsh[w][d < 0 ? 0 : d];
      A[s][j] = (d < 0) ? 0.f : hv;
    }
  }
  // boundary coefficients per C/D VGPR r (row = r + roff):
  //   x[row] += h[row+1]*x(-1) + (-a1*h[row])*x(-2)
  float hp[8], hq[8];
#pragma unroll
  for (int r = 0; r < 8; ++r) {
    hp[r] = hsh[w][r + roff + 1];
    hq[r] = -a1 * hsh[w][r + roff];
  }

  // initial states per column batch
  const int bcol = bbase + m;
  float x1 = y0[(((size_t)bcol * Osz + o) * Isz + i) * NAt + 0];
  float x2 = y0[(((size_t)bcol * Osz + o) * Isz + i) * NAt + 1];

  // ---- TDM descriptor (group1 is loop-invariant) ----
  const uint64_t ubase = (uint64_t)(uintptr_t)u_in;
  const uint32_t lds0 = (uint32_t)(uintptr_t)&ubuf[0][0];
  const uint32_t lds1 = (uint32_t)(uintptr_t)&ubuf[1][0];
  i32x8 g1v;
  g1v[0] = 0x00020000;                       // data_size = 4B
  g1v[1] = 0;                                // tensor_dim0[15:0]<<16 (T*I=131072 -> 0)
  g1v[2] = (int)((131072u >> 16) | (32u << 16)); // tensor_dim0 hi | tensor_dim1=32
  g1v[3] = (int)(144u << 16);                // tile_dim0 = 144 elements (18*8 f32)
  g1v[4] = 16;                               // tile_dim1 = 16 batches
  g1v[5] = 131072;                           // tensor_dim0_stride = T*I elements
  g1v[6] = 0;
  g1v[7] = 0;
  const i32x4 zz4 = {0, 0, 0, 0};
  const i32x8 zz8 = {0, 0, 0, 0, 0, 0, 0, 0};

  for (int n = 0; n < NT; ++n) {
    // ---- pipeline: wave 0 issues TDM for tile n+1, waits for tile n ----
    if (w == 0) {
      if (n + 1 < NT) {
        uint64_t ga = ubase +
            (((uint64_t)bbase * Tsz + (uint64_t)(16 * (n + 1) - 2)) * Isz) * 4u;
        u32x4 g0v;
        g0v[0] = 1u;                                          // count=1
        g0v[1] = ((n + 1) & 1) ? lds1 : lds0;                 // lds_addr
        g0v[2] = (uint32_t)(ga & 0xFFFFFFFFu);                // global_addr lo
        g0v[3] = (uint32_t)((ga >> 32) & 0x1FFFFFFu) | 0x80000000u; // hi | type=2
        __builtin_amdgcn_tensor_load_to_lds(g0v, g1v, zz4, zz4, zz8, 0);
        __builtin_amdgcn_s_wait_tensorcnt(1);                 // tile n complete
      } else {
        __builtin_amdgcn_s_wait_tensorcnt(0);
      }
    }
    __syncthreads();

    const float* ub = ubuf[n & 1];

    // ---- FIR tile (B operand) from LDS + 4x WMMA f32 16x16x4 ----
    v8f acc = {0.f, 0.f, 0.f, 0.f, 0.f, 0.f, 0.f, 0.f};
#pragma unroll
    for (int s = 0; s < 4; ++s) {
      v2f Bv;
#pragma unroll
      for (int j = 0; j < 2; ++j) {
        int K = 4 * s + j + khalf;                       // time row within tile
        const float* up = &ub[m * (18 * Isz) + K * Isz + i];
        // fir[t0+K] = b0*u[t0+K] + b1*u[t0+K-1] + b2*u[t0+K-2]
        Bv[j] = b2 * up[0] + b1 * up[Isz] + b0 * up[2 * Isz];
      }
      acc = __builtin_amdgcn_wmma_f32_16x16x4_f32(
          false, A[s], false, Bv, (short)0, acc, false, false);
    }

    // ---- rank-2 boundary correction from previous tile's exit states ----
#pragma unroll
    for (int r = 0; r < 8; ++r) acc[r] += hp[r] * x1 + hq[r] * x2;

    // exit states: rows 15 / 14 live in lanes 16..31 of acc[7] / acc[6]
    x2 = __shfl(acc[6], 16 + m, 32);
    x1 = __shfl(acc[7], 16 + m, 32);

    // ---- stage per-wave tiles, reduce over the 8 input channels, store y ----
#pragma unroll
    for (int r = 0; r < 8; ++r)
      ystage[(w * TILE + r + roff) * BG + m] = acc[r];
    __syncthreads();

    {
      int row = tid & 15;       // time within tile
      int c2  = tid >> 4;       // batch column
      float ssum = 0.f;
#pragma unroll
      for (int ww = 0; ww < 8; ++ww)
        ssum += ystage[(ww * TILE + row) * BG + c2];
      yout[(((size_t)(bbase + c2)) * Tsz + (size_t)(n * TILE + row)) * Osz + o] = ssum;
    }
  }
}

extern "C" void kernel_launch(void* const* d_in, const int* in_sizes, int n_in,
                              void* d_out, int out_size, void* d_ws, size_t ws_size,
                              hipStream_t stream) {
  const float* bco  = (const float*)d_in[0];
  const float* aco  = (const float*)d_in[1];
  const float* u_in = (const float*)d_in[2];
  const float* y0   = (const float*)d_in[3];
  const float* u0   = (const float*)d_in[4];
  float* yout = (float*)d_out;

  dim3 grid(NGRP * Osz);   // 16 workgroups: (o, batch-group)
  dim3 block(256);         // 8 waves = 8 input channels
  hipLaunchKernelGGL(mimo_iir_kernel, grid, block, 0, stream,
                     bco, aco, u_in, y0, u0, yout);
}